// ARIMAModel_73083163508835
// MI455X (gfx1250) — compile-verified
//
#include <hip/hip_runtime.h>
#include <hip/hip_bf16.h>

// ARIMA window: out[b,n,i,f] = sum_a (ar[a,f]+ma[a,f]) * x[b,n,i-8+a,f], i>=8 else 0.
// Memory-bound (AI ~2 flop/B). LDS-tiled streaming + V_WMMA_F32_16X16X4_F32 compute.

typedef __attribute__((ext_vector_type(2))) float v2f;
typedef __attribute__((ext_vector_type(8))) float v8f;

#define S_DIM   4096
#define F_DIM   8
#define K_WIN   8
#define CHUNK   128
#define ROWS_IN (CHUNK + K_WIN)   // 136 staged rows (8-row halo)

__global__ __launch_bounds__(256, 2)
void arima_wmma_kernel(const float* __restrict__ x,
                       const float* __restrict__ ar,
                       const float* __restrict__ ma,
                       float* __restrict__ out)
{
    const int tid            = threadIdx.x;
    const int chunks_per_bn  = S_DIM / CHUNK;            // 32
    const int bn             = blockIdx.x / chunks_per_bn;
    const int chunk          = blockIdx.x % chunks_per_bn;
    const long long base     = (long long)bn * (S_DIM * F_DIM);
    const int srow0          = chunk * CHUNK;            // first output row of chunk

    __shared__ __align__(16) float lds_in [ROWS_IN * F_DIM];   // 4352 B
    __shared__ __align__(16) float lds_w  [K_WIN * F_DIM];     // combined weights
    __shared__ __align__(16) float lds_out[8 * 16 * F_DIM];    // per-wave D staging

    // Combined weights W[a,f] = ar[a,f] + ma[a,f]
    if (tid < K_WIN * F_DIM) lds_w[tid] = ar[tid] + ma[tid];

    // Stage rows [srow0-8, srow0+128) as float4 (2 per 32B row); zero-fill halo at i<0.
    const int NF4 = ROWS_IN * 2;  // 272
    for (int f4 = tid; f4 < NF4; f4 += 256) {
        const int r    = f4 >> 1;
        const int h    = f4 & 1;
        const int grow = srow0 - K_WIN + r;              // global row; max = srow0+127 < S
        float4 v = make_float4(0.f, 0.f, 0.f, 0.f);
        if (grow >= 0)
            v = *(const float4*)(x + base + (long long)grow * F_DIM + h * 4);
        *(float4*)(lds_in + f4 * 4) = v;
    }
    __syncthreads();

    // ---- per-wave 16-row tile via V_WMMA_F32_16X16X4_F32 ----
    const int wave = tid >> 5;
    const int lane = tid & 31;
    const int lm   = lane & 15;      // M index for A rows / N index for B,D cols
    const int hi   = lane >> 4;      // 0: K=0,1  1: K=2,3 (A & B VGPR layout halves)
    const int i0   = wave * 16;      // local output row of this tile

    v8f c = {0.f, 0.f, 0.f, 0.f, 0.f, 0.f, 0.f, 0.f};
    const int k0 = 2 * hi;
    const float wcol_base0 = 0.f;    // keep EXEC uniform: no divergence in this region
    (void)wcol_base0;

    #pragma unroll
    for (int a = 0; a < K_WIN; ++a) {
        // W[a, n] for this lane's output column n=lm (only n<8 meaningful; b-select zeroes rest)
        const float wn = lds_w[a * F_DIM + (lm & 7)];
        #pragma unroll
        for (int fb = 0; fb < F_DIM; fb += 4) {
            // A[m,k] = x_row(i0+m-8+a)[fb+k]  -> LDS row (i0 + m + a)
            const int row = i0 + lm + a;
            v2f av = *(const v2f*)(lds_in + row * F_DIM + fb + k0);
            // B[k,n] = W[a,n] * delta(fb+k, n)  (diagonal-expanded depthwise weights)
            v2f bv;
            bv.x = (lm == fb + k0    ) ? wn : 0.f;
            bv.y = (lm == fb + k0 + 1) ? wn : 0.f;
            // 8 args: (neg_a, A, neg_b, B, c_mod, C, reuse_a, reuse_b)
            c = __builtin_amdgcn_wmma_f32_16x16x4_f32(false, av, false, bv,
                                                      (short)0, c, false, false);
        }
    }

    // D layout: VGPR v, lanes 0-15 -> (M=v, N=lane); lanes 16-31 -> (M=v+8, N=lane-16).
    // Transpose through LDS for coalesced 128-bit stores.
    float* wout = lds_out + wave * (16 * F_DIM);
    if (lm < F_DIM) {
        #pragma unroll
        for (int v = 0; v < 8; ++v)
            wout[(v + 8 * hi) * F_DIM + lm] = c[v];
    }
    __syncthreads();

    // 512 B per wave, fully coalesced; force rows i<8 of the series to zero.
    {
        const int gi = srow0 + i0 + (lane >> 1);   // output row owning this lane's float4
        float4 v = *(const float4*)(lds_out + wave * (16 * F_DIM) + lane * 4);
        if (gi < K_WIN) v = make_float4(0.f, 0.f, 0.f, 0.f);
        *(float4*)(out + base + (long long)(srow0 + i0) * F_DIM + lane * 4) = v;
    }
}

extern "C" void kernel_launch(void* const* d_in, const int* in_sizes, int n_in,
                              void* d_out, int out_size, void* d_ws, size_t ws_size,
                              hipStream_t stream) {
    const float* x  = (const float*)d_in[0];
    const float* ar = (const float*)d_in[1];
    const float* ma = (const float*)d_in[2];
    float* out = (float*)d_out;

    const int bn_total = in_sizes[0] / (S_DIM * F_DIM);        // B*N = 800
    const int blocks   = bn_total * (S_DIM / CHUNK);           // 25600
    arima_wmma_kernel<<<blocks, 256, 0, stream>>>(x, ar, ma, out);
}